// HeteroNewsCompanyGNN_12412455486145
// MI455X (gfx1250) — compile-verified
//
#include <hip/hip_runtime.h>
#include <hip/hip_bf16.h>

// CDNA5 fp32 WMMA types: 16x16x4_f32 -> A,B = 2 VGPRs (v2f), C/D = 8 VGPRs (v8f)
typedef __attribute__((ext_vector_type(2))) float v2f;
typedef __attribute__((ext_vector_type(8))) float v8f;

#define NN 100000
#define NC 5000
#define HDIM 64
#define ES 3200000
#define EM 1000000
#define ER 160000

// ---------------------------------------------------------------------------
__global__ __launch_bounds__(256) void k_zero(float* __restrict__ p, long n4) {
    long i = (long)blockIdx.x * blockDim.x + threadIdx.x;
    if (i < n4) ((float4*)p)[i] = make_float4(0.f, 0.f, 0.f, 0.f);
}

// ---------------------------------------------------------------------------
// Pack weight W[K x N] into K-pair-interleaved form, zero-padded to K%4==0:
// Wp[k2 * N + col] = { W[2*k2][col], W[2*k2+1][col] }  -> B loads become b64.
__global__ __launch_bounds__(256) void k_packW(const float* __restrict__ W,
                                               v2f* __restrict__ Wp,
                                               int K, int N) {
    const int Kp2 = ((K + 3) & ~3) >> 1;        // padded half-rows
    const long n = (long)Kp2 * N;
    long i = (long)blockIdx.x * blockDim.x + threadIdx.x;
    if (i >= n) return;
    int k2 = (int)(i / N), col = (int)(i % N);
    int k = 2 * k2;
    v2f v;
    v.x = (k     < K) ? W[(long)k * N + col]       : 0.f;
    v.y = (k + 1 < K) ? W[(long)(k + 1) * N + col] : 0.f;
    Wp[i] = v;
}

// ---------------------------------------------------------------------------
// Edge scatter: 16 threads per edge; each moves a float4 of the 64-float row.
__global__ __launch_bounds__(256) void k_scatter(const int* __restrict__ src,
                                                 const int* __restrict__ dst,
                                                 const float* __restrict__ x,
                                                 float* __restrict__ agg,
                                                 float* __restrict__ cnt,
                                                 long nE) {
    long gid = (long)blockIdx.x * blockDim.x + threadIdx.x;
    long e = gid >> 4;
    int c = (int)(gid & 15) * 4;
    if (e >= nE) return;
    int s = src[e], d = dst[e];
    const float4 v = *(const float4*)(x + (long)s * HDIM + c);
    float* ap = agg + (long)d * HDIM + c;
    atomicAdd(ap + 0, v.x);
    atomicAdd(ap + 1, v.y);
    atomicAdd(ap + 2, v.z);
    atomicAdd(ap + 3, v.w);
    if ((gid & 15) == 0) atomicAdd(cnt + d, 1.0f);
}

// ---------------------------------------------------------------------------
__global__ __launch_bounds__(256) void k_scale_rows(float* __restrict__ agg,
                                                    const float* __restrict__ cnt,
                                                    int M) {
    long gid = (long)blockIdx.x * blockDim.x + threadIdx.x;
    long row = gid >> 4;
    int c = (int)(gid & 15) * 4;
    if (row >= M) return;
    float inv = 1.0f / fmaxf(cnt[row], 1.0f);
    float4* p = (float4*)(agg + row * HDIM + c);
    float4 v = *p;
    v.x *= inv; v.y *= inv; v.z *= inv; v.w *= inv;
    *p = v;
}

// ---------------------------------------------------------------------------
// WMMA fp32 GEMM body. FULL = all 16 rows in range. VECA = A rows 8B-aligned
// (K even) so A loads are b64. B always b64 from the packed weight.
template<int NT, bool FULL, bool VECA>
__device__ __forceinline__ void gemm_body(v8f (&c)[NT],
                                          const float* __restrict__ A,
                                          const v2f* __restrict__ Wp,
                                          int M, int K, int N,
                                          int mbase, int lane) {
    const int row  = mbase + (lane & 15);
    const int koff = (lane >= 16) ? 2 : 0;
    const int col0 = lane & 15;
    const bool rok = FULL || (row < M);
    const float* __restrict__ Arow = A + (long)row * K;
    const int Kmain = K & ~3;

    #pragma unroll 4
    for (int k = 0; k < Kmain; k += 4) {
        const int kk = k + koff;
        v2f a;
        if (rok) {
            if (VECA) {
                a = *(const v2f*)(Arow + kk);
            } else {
                a.x = Arow[kk];
                a.y = Arow[kk + 1];
            }
        } else {
            a.x = 0.f; a.y = 0.f;
        }
        #pragma unroll
        for (int nt = 0; nt < NT; ++nt) {
            v2f b = Wp[(long)(kk >> 1) * N + nt * 16 + col0];
            c[nt] = __builtin_amdgcn_wmma_f32_16x16x4_f32(
                false, a, false, b, (short)0, c[nt], false, false);
        }
    }
    if (Kmain < K) {               // K tail (K=385): A guarded, B padded (safe)
        const int kk = Kmain + koff;
        v2f a;
        a.x = (rok && kk     < K) ? Arow[kk]     : 0.f;
        a.y = (rok && kk + 1 < K) ? Arow[kk + 1] : 0.f;
        #pragma unroll
        for (int nt = 0; nt < NT; ++nt) {
            v2f b = Wp[(long)(kk >> 1) * N + nt * 16 + col0];
            c[nt] = __builtin_amdgcn_wmma_f32_16x16x4_f32(
                false, a, false, b, (short)0, c[nt], false, false);
        }
    }
}

// out[M x N] (=|+=) A1@W1 (+ A2@W2) + bias, optional relu. N = NT*16.
template<int NT, bool VECA1>
__global__ __launch_bounds__(256) void k_gemm(const float* __restrict__ A1,
                                              const v2f* __restrict__ W1p,
                                              const float* __restrict__ bias,
                                              const float* __restrict__ A2,
                                              const v2f* __restrict__ W2p,
                                              float* __restrict__ out,
                                              int M, int K1, int K2,
                                              int do_relu, int accum) {
    const int N = NT * 16;
    const int lane = threadIdx.x & 31;
    const int wv = threadIdx.x >> 5;
    const int mbase = (blockIdx.x * 8 + wv) * 16;
    if (mbase >= M) return;

    v8f c[NT];
    #pragma unroll
    for (int t = 0; t < NT; ++t)
        c[t] = (v8f){0.f, 0.f, 0.f, 0.f, 0.f, 0.f, 0.f, 0.f};

    const bool full = (mbase + 16 <= M);
    if (full) {
        gemm_body<NT, true, VECA1>(c, A1, W1p, M, K1, N, mbase, lane);
        if (A2) gemm_body<NT, true, true>(c, A2, W2p, M, K2, N, mbase, lane);
    } else {
        gemm_body<NT, false, VECA1>(c, A1, W1p, M, K1, N, mbase, lane);
        if (A2) gemm_body<NT, false, true>(c, A2, W2p, M, K2, N, mbase, lane);
    }

    // C/D layout: VGPR r -> row = mbase + r + (lane>=16 ? 8 : 0), col = nt*16 + (lane&15)
    const int col0 = lane & 15;
    const int rofs = (lane >= 16) ? 8 : 0;
    #pragma unroll
    for (int nt = 0; nt < NT; ++nt) {
        const int col = nt * 16 + col0;
        const float bb = bias ? bias[col] : 0.f;
        #pragma unroll
        for (int r = 0; r < 8; ++r) {
            const int row = mbase + r + rofs;
            if (full || row < M) {
                const long idx = (long)row * N + col;
                float v = c[nt][r] + bb;
                if (accum) v += out[idx];
                if (do_relu) v = fmaxf(v, 0.f);
                out[idx] = v;
            }
        }
    }
}

// ---------------------------------------------------------------------------
// Fused (residual add) + relu + LayerNorm over rows of 64. One wave per row.
__global__ void k_ln(float* __restrict__ out, const float* __restrict__ in,
                     const float* __restrict__ res,
                     const float* __restrict__ g, const float* __restrict__ b,
                     int M) {
    int lane = threadIdx.x;
    int row = blockIdx.x * blockDim.y + threadIdx.y;
    if (row >= M) return;
    long base = (long)row * HDIM;
    float v0 = in[base + lane];
    float v1 = in[base + lane + 32];
    if (res) { v0 += res[base + lane]; v1 += res[base + lane + 32]; }
    v0 = fmaxf(v0, 0.f);
    v1 = fmaxf(v1, 0.f);
    float s = v0 + v1;
    for (int off = 16; off > 0; off >>= 1) s += __shfl_xor(s, off, 32);
    float mean = s * (1.0f / 64.0f);
    float d0 = v0 - mean, d1 = v1 - mean;
    float q = d0 * d0 + d1 * d1;
    for (int off = 16; off > 0; off >>= 1) q += __shfl_xor(q, off, 32);
    float rstd = rsqrtf(q * (1.0f / 64.0f) + 1e-5f);
    out[base + lane]      = d0 * rstd * g[lane]      + b[lane];
    out[base + lane + 32] = d1 * rstd * g[lane + 32] + b[lane + 32];
}

// ---------------------------------------------------------------------------
__global__ __launch_bounds__(256) void k_addrelu(float* __restrict__ o,
                                                 const float* __restrict__ a,
                                                 const float* __restrict__ b,
                                                 long n) {
    long i = (long)blockIdx.x * blockDim.x + threadIdx.x;
    if (i < n) o[i] = fmaxf(a[i] + b[i], 0.f);
}

// ---------------------------------------------------------------------------
__global__ __launch_bounds__(256) void k_final(const float* __restrict__ h,
                                               const float* __restrict__ W2,
                                               const float* __restrict__ b2,
                                               float* __restrict__ out, int M) {
    int i = blockIdx.x * blockDim.x + threadIdx.x;
    if (i >= M) return;
    float acc = b2[0];
    const float* hr = h + (long)i * 32;
    #pragma unroll
    for (int j = 0; j < 32; ++j) acc += hr[j] * W2[j];
    out[i] = acc;
}

// ---------------------------------------------------------------------------
extern "C" void kernel_launch(void* const* d_in, const int* in_sizes, int n_in,
                              void* d_out, int out_size, void* d_ws, size_t ws_size,
                              hipStream_t stream) {
    const float* x_news    = (const float*)d_in[0];
    const float* x_company = (const float*)d_in[1];
    const int*   edge_sim  = (const int*)d_in[2];   // [2, ES]
    const int*   men_src   = (const int*)d_in[3];
    const int*   men_dst   = (const int*)d_in[4];
    const int*   edge_rel  = (const int*)d_in[5];   // [2, ER]
    const float* nw_W = (const float*)d_in[6];
    const float* nw_b = (const float*)d_in[7];
    const float* cw_W = (const float*)d_in[8];
    const float* cw_b = (const float*)d_in[9];
    const float* n1n_g = (const float*)d_in[10];
    const float* n1n_b = (const float*)d_in[11];
    const float* n1c_g = (const float*)d_in[12];
    const float* n1c_b = (const float*)d_in[13];
    const float* c1_Wl = (const float*)d_in[14];    // [3,64,64]
    const float* c1_bl = (const float*)d_in[15];    // [3,64]
    const float* c1_Wr = (const float*)d_in[16];
    const float* c2_Wl = (const float*)d_in[17];
    const float* c2_bl = (const float*)d_in[18];
    const float* c2_Wr = (const float*)d_in[19];
    const float* cls_W1 = (const float*)d_in[20];   // [64,32]
    const float* cls_b1 = (const float*)d_in[21];
    const float* cls_W2 = (const float*)d_in[22];   // [32,1]
    const float* cls_b2 = (const float*)d_in[23];
    float* out = (float*)d_out;

    // ---- workspace carve-up (floats) ----
    float* ws = (float*)d_ws;
    long o = 0;
    float* xn0  = ws + o; o += (long)NN * HDIM;
    float* xn1  = ws + o; o += (long)NN * HDIM;
    float* aggn = ws + o; o += (long)NN * HDIM;     // [aggn|cntn] contiguous zero region
    float* cntn = ws + o; o += NN;
    float* aggcM = ws + o; o += (long)NC * HDIM;    // [aggcM|cntcM|aggcR|cntcR] contiguous
    float* cntcM = ws + o; o += NC;
    float* aggcR = ws + o; o += (long)NC * HDIM;
    float* cntcR = ws + o; o += NC;
    float* xc0  = ws + o; o += (long)NC * HDIM;
    float* xc1  = ws + o; o += (long)NC * HDIM;
    float* ocb  = ws + o; o += (long)NC * HDIM;
    float* comp = ws + o; o += (long)NC * HDIM;
    float* hbuf = ws + o; o += (long)NC * 32;
    // packed weights (v2f = 2 floats each)
    v2f* wp_nw  = (v2f*)(ws + o); o += 2L * (388 / 2) * 64;   // K=385 -> pad 388
    v2f* wp_cw  = (v2f*)(ws + o); o += 2L * (24 / 2) * 64;
    v2f* wp_c1  = (v2f*)(ws + o); o += 2L * 6 * 32 * 64;      // Wl0,Wr0,Wl1,Wr1,Wl2,Wr2
    v2f* wp_c2  = (v2f*)(ws + o); o += 2L * 4 * 32 * 64;      // Wl1,Wr1,Wl2,Wr2
    v2f* wp_cls = (v2f*)(ws + o); o += 2L * 32 * 32;

    const long zn_n = (long)NN * HDIM + NN;                 // news agg+cnt
    const long zn_c = 2L * ((long)NC * HDIM + NC);          // company aggs+cnts

    dim3 blk(256);
    auto blocks = [](long n) { return (unsigned)((n + 255) / 256); };
    int tilesNN = (NN + 15) / 16, tilesNC = (NC + 15) / 16;
    unsigned gemmBlkNN = (tilesNN + 7) / 8, gemmBlkNC = (tilesNC + 7) / 8;

    // 0) pack all weights into K-pair-interleaved (b64-loadable) form
    k_packW<<<blocks((388L / 2) * 64), blk, 0, stream>>>(nw_W, wp_nw, 385, 64);
    k_packW<<<blocks((24L / 2) * 64), blk, 0, stream>>>(cw_W, wp_cw, 24, 64);
    for (int r = 0; r < 3; ++r) {
        k_packW<<<blocks(32L * 64), blk, 0, stream>>>(c1_Wl + r * 4096,
                                                      wp_c1 + (2 * r) * 2048, 64, 64);
        k_packW<<<blocks(32L * 64), blk, 0, stream>>>(c1_Wr + r * 4096,
                                                      wp_c1 + (2 * r + 1) * 2048, 64, 64);
    }
    for (int r = 1; r < 3; ++r) {
        k_packW<<<blocks(32L * 64), blk, 0, stream>>>(c2_Wl + r * 4096,
                                                      wp_c2 + (2 * (r - 1)) * 2048, 64, 64);
        k_packW<<<blocks(32L * 64), blk, 0, stream>>>(c2_Wr + r * 4096,
                                                      wp_c2 + (2 * (r - 1) + 1) * 2048, 64, 64);
    }
    k_packW<<<blocks(32L * 32), blk, 0, stream>>>(cls_W1, wp_cls, 64, 32);

    // 1) feature transforms (WMMA fp32), relu
    k_gemm<4, false><<<gemmBlkNN, blk, 0, stream>>>(x_news, wp_nw, nw_b,
                                                    nullptr, nullptr,
                                                    xn0, NN, 385, 0, 1, 0);
    k_gemm<4, true><<<gemmBlkNC, blk, 0, stream>>>(x_company, wp_cw, cw_b,
                                                   nullptr, nullptr,
                                                   xc0, NC, 24, 0, 1, 0);

    // 2) layer-1 aggregation
    k_zero<<<blocks(zn_n / 4), blk, 0, stream>>>(aggn, zn_n / 4);
    k_zero<<<blocks(zn_c / 4), blk, 0, stream>>>(aggcM, zn_c / 4);
    k_scatter<<<blocks((long)ES * 16), blk, 0, stream>>>(edge_sim, edge_sim + ES,
                                                         xn0, aggn, cntn, ES);
    k_scatter<<<blocks((long)EM * 16), blk, 0, stream>>>(men_src, men_dst,
                                                         xn0, aggcM, cntcM, EM);
    k_scatter<<<blocks((long)ER * 16), blk, 0, stream>>>(edge_rel, edge_rel + ER,
                                                         xc0, aggcR, cntcR, ER);
    k_scale_rows<<<blocks((long)NN * 16), blk, 0, stream>>>(aggn, cntn, NN);
    k_scale_rows<<<blocks((long)NC * 16), blk, 0, stream>>>(aggcM, cntcM, NC);
    k_scale_rows<<<blocks((long)NC * 16), blk, 0, stream>>>(aggcR, cntcR, NC);

    // 3) layer-1 linear: on = aggn@Wl0 + bl0 + xn0@Wr0 ; oc = two fused SAGEs
    k_gemm<4, true><<<gemmBlkNN, blk, 0, stream>>>(aggn, wp_c1 + 0 * 2048,
                                                   c1_bl + 0 * 64,
                                                   xn0, wp_c1 + 1 * 2048,
                                                   xn1, NN, 64, 64, 0, 0);
    k_gemm<4, true><<<gemmBlkNC, blk, 0, stream>>>(aggcM, wp_c1 + 2 * 2048,
                                                   c1_bl + 1 * 64,
                                                   xc0, wp_c1 + 3 * 2048,
                                                   ocb, NC, 64, 64, 0, 0);
    k_gemm<4, true><<<gemmBlkNC, blk, 0, stream>>>(aggcR, wp_c1 + 4 * 2048,
                                                   c1_bl + 2 * 64,
                                                   xc0, wp_c1 + 5 * 2048,
                                                   ocb, NC, 64, 64, 0, 1);   // accumulate

    // 4) norms: xn1 = LN(relu(on)) in place; xc1 = LN(relu(oc + xc0))
    dim3 lnb(32, 8);
    k_ln<<<(NN + 7) / 8, lnb, 0, stream>>>(xn1, xn1, nullptr, n1n_g, n1n_b, NN);
    k_ln<<<(NC + 7) / 8, lnb, 0, stream>>>(xc1, ocb, xc0, n1c_g, n1c_b, NC);

    // 5) layer-2: only company outputs are live (on2 is dead code)
    k_zero<<<blocks(zn_c / 4), blk, 0, stream>>>(aggcM, zn_c / 4);
    k_scatter<<<blocks((long)EM * 16), blk, 0, stream>>>(men_src, men_dst,
                                                         xn1, aggcM, cntcM, EM);
    k_scatter<<<blocks((long)ER * 16), blk, 0, stream>>>(edge_rel, edge_rel + ER,
                                                         xc1, aggcR, cntcR, ER);
    k_scale_rows<<<blocks((long)NC * 16), blk, 0, stream>>>(aggcM, cntcM, NC);
    k_scale_rows<<<blocks((long)NC * 16), blk, 0, stream>>>(aggcR, cntcR, NC);
    k_gemm<4, true><<<gemmBlkNC, blk, 0, stream>>>(aggcM, wp_c2 + 0 * 2048,
                                                   c2_bl + 1 * 64,
                                                   xc1, wp_c2 + 1 * 2048,
                                                   ocb, NC, 64, 64, 0, 0);
    k_gemm<4, true><<<gemmBlkNC, blk, 0, stream>>>(aggcR, wp_c2 + 2 * 2048,
                                                   c2_bl + 2 * 64,
                                                   xc1, wp_c2 + 3 * 2048,
                                                   ocb, NC, 64, 64, 0, 1);

    // 6) head: comp = relu(oc2 + xc1); h = relu(comp@W1+b1); out = h@W2+b2
    k_addrelu<<<blocks((long)NC * HDIM), blk, 0, stream>>>(comp, ocb, xc1,
                                                           (long)NC * HDIM);
    k_gemm<2, true><<<gemmBlkNC, blk, 0, stream>>>(comp, wp_cls, cls_b1,
                                                   nullptr, nullptr,
                                                   hbuf, NC, 64, 0, 1, 0);
    k_final<<<blocks(NC), blk, 0, stream>>>(hbuf, cls_W2, cls_b2, out, NC);
}